// NMSWrapper_60464549593386
// MI455X (gfx1250) — compile-verified
//
#include <hip/hip_runtime.h>
#include <stdint.h>

#define NMS_N    4096
#define NMS_T    1024
#define NMS_PER  (NMS_N / NMS_T)   // 4 elements per thread
#define SCORE_TH 0.3f
#define IOU_TH   0.5f

// One workgroup (32 waves, wave32) per batch image. Batch state is fully
// LDS-resident on one WGP: keys 32KB + staged scores 16KB + active 4KB = 52KB.
__global__ __launch_bounds__(NMS_T) void nms_gfx1250_kernel(
    const float* __restrict__ boxes,   // [B, N, 4] xyxy
    const float* __restrict__ scores,  // [B, N]
    float* __restrict__ out)           // [B, N, 5]
{
    __shared__ unsigned long long keys[NMS_N];  // (~sortable(score))<<32 | idx
    __shared__ float              sscore[NMS_N];
    __shared__ unsigned char      act[NMS_N];

    const int b   = blockIdx.x;
    const int tid = threadIdx.x;
    const float* bbase = boxes  + (size_t)b * NMS_N * 4;
    const float* sbase = scores + (size_t)b * NMS_N;

    // ---- Phase 0: CDNA5 async global->LDS staging of scores + init flags ----
    #pragma unroll
    for (int m = 0; m < NMS_PER; ++m) {
        const int e = tid + m * NMS_T;
        // Low 32 bits of a generic LDS pointer are the workgroup-relative LDS
        // byte offset (flat aperture layout: addr[31:0] = LDS offset).
        uint32_t laddr = (uint32_t)(uintptr_t)(&sscore[e]);
        const float* g = sbase + e;
        asm volatile("global_load_async_to_lds_b32 %0, %1, off"
                     :: "v"(laddr), "v"(g) : "memory");
        act[e] = (unsigned char)1;
    }
    asm volatile("s_wait_asynccnt 0" ::: "memory");  // ASYNCcnt == 0
    __syncthreads();

    // ---- Phase 1: build sort keys (descending score, ascending idx) ----
    #pragma unroll
    for (int m = 0; m < NMS_PER; ++m) {
        const int e = tid + m * NMS_T;
        uint32_t ub = __float_as_uint(sscore[e]);
        // float -> monotonically-increasing uint
        uint32_t s  = ub ^ ((ub >> 31) ? 0xFFFFFFFFu : 0x80000000u);
        keys[e] = ((unsigned long long)(uint32_t)(~s) << 32) | (uint32_t)e;
    }

    // ---- Phase 2: bitonic sort of 4096 u64 keys in LDS (ascending) ----
    for (unsigned k = 2; k <= NMS_N; k <<= 1) {
        for (unsigned j = k >> 1; j > 0; j >>= 1) {
            __syncthreads();
            #pragma unroll
            for (int m = 0; m < NMS_PER; ++m) {
                const unsigned e  = (unsigned)tid + (unsigned)m * NMS_T;
                const unsigned ix = e ^ j;
                if (ix > e) {
                    unsigned long long a = keys[e];
                    unsigned long long c = keys[ix];
                    const bool up = ((e & k) == 0);
                    if ((a > c) == up) { keys[e] = c; keys[ix] = a; }
                }
            }
        }
    }
    __syncthreads();

    // ---- Phase 3: gather this thread's sorted boxes into VGPRs ----
    float jx1[NMS_PER], jy1[NMS_PER], jx2[NMS_PER], jy2[NMS_PER];
    float jar[NMS_PER], jsc[NMS_PER];
    #pragma unroll
    for (int m = 0; m < NMS_PER; ++m) {
        const int e = tid + m * NMS_T;
        const unsigned idx = (unsigned)(keys[e] & 0xFFFFFFFFull);
        const float4 bx = *reinterpret_cast<const float4*>(bbase + (size_t)idx * 4);
        jx1[m] = bx.x; jy1[m] = bx.y; jx2[m] = bx.z; jy2[m] = bx.w;
        jar[m] = (bx.z - bx.x) * (bx.w - bx.y);
        jsc[m] = sscore[idx];           // LDS gather (staged async above)
    }

    // ---- Phase 4: serial greedy suppression, parallel over candidates ----
    for (int i = 0; i < NMS_N; ++i) {
        __syncthreads();                 // make iteration i-1 writes visible
        // All lanes read the same LDS locations -> block-uniform branch.
        if (!act[i]) continue;
        const unsigned long long ki = keys[i];
        const unsigned idxi = (unsigned)(ki & 0xFFFFFFFFull);
        const float si = sscore[idxi];
        if (!(si > SCORE_TH)) continue;  // invalid boxes never suppress

        const float4 bi = *reinterpret_cast<const float4*>(bbase + (size_t)idxi * 4);
        const float aarea = (bi.z - bi.x) * (bi.w - bi.y);

        #pragma unroll
        for (int m = 0; m < NMS_PER; ++m) {
            const int jj = tid + m * NMS_T;
            if (jj > i) {                // only suppress lower-scored boxes
                float lx = fmaxf(bi.x, jx1[m]);
                float ly = fmaxf(bi.y, jy1[m]);
                float rx = fminf(bi.z, jx2[m]);
                float ry = fminf(bi.w, jy2[m]);
                float iw = fmaxf(rx - lx, 0.0f);
                float ih = fmaxf(ry - ly, 0.0f);
                float inter = iw * ih;
                float uni   = aarea + jar[m] - inter;
                float iou   = inter / fmaxf(uni, 1e-9f);
                if (iou > IOU_TH) act[jj] = (unsigned char)0;
            }
        }
    }
    __syncthreads();

    // ---- Phase 5: write [B,N,5] = (sorted box, score) * keep ----
    #pragma unroll
    for (int m = 0; m < NMS_PER; ++m) {
        const int e = tid + m * NMS_T;
        const float keep = (act[e] && (jsc[m] > SCORE_TH)) ? 1.0f : 0.0f;
        float* o = out + ((size_t)b * NMS_N + e) * 5;
        o[0] = jx1[m] * keep;
        o[1] = jy1[m] * keep;
        o[2] = jx2[m] * keep;
        o[3] = jy2[m] * keep;
        o[4] = jsc[m] * keep;
    }
}

extern "C" void kernel_launch(void* const* d_in, const int* in_sizes, int n_in,
                              void* d_out, int out_size, void* d_ws, size_t ws_size,
                              hipStream_t stream) {
    (void)n_in; (void)out_size; (void)d_ws; (void)ws_size;
    const float* boxes  = (const float*)d_in[0];   // [B,N,4]
    const float* scores = (const float*)d_in[1];   // [B,N]
    float* out = (float*)d_out;                    // [B,N,5]
    const int B = in_sizes[1] / NMS_N;             // = 8
    nms_gfx1250_kernel<<<B, NMS_T, 0, stream>>>(boxes, scores, out);
}